// GNNLayer_28329604285092
// MI455X (gfx1250) — compile-verified
//
#include <hip/hip_runtime.h>
#include <math.h>

#define DIM 128

typedef float v2f __attribute__((ext_vector_type(2)));
typedef float v8f __attribute__((ext_vector_type(8)));

// ---------------------------------------------------------------------------
// 0. per-call reset of segment-softmax state
// ---------------------------------------------------------------------------
__global__ void init_nodes(float* __restrict__ smax, float* __restrict__ denom, int n) {
    int i = blockIdx.x * blockDim.x + threadIdx.x;
    if (i < n) { smax[i] = -INFINITY; denom[i] = 0.0f; }
}

// ---------------------------------------------------------------------------
// 1. fused node-level GEMMs via V_WMMA_F32_16X16X4_F32
//    msg = h@Ww^T + Wb ; q = h@qw^T + qb ; k = h@kw^T + kb ;
//    out = h@WSw^T + WSb   (scatter later accumulates h_agg on top of out)
//    block = 256 thr (8 waves); block handles one 16-row node tile;
//    wave w handles the 16-col tile n0 = 16*w for all 4 weight matrices.
// ---------------------------------------------------------------------------
__global__ __launch_bounds__(256) void node_gemm4(
    const float* __restrict__ h,
    const float* __restrict__ Ww,  const float* __restrict__ Wb,
    const float* __restrict__ WSw, const float* __restrict__ WSb,
    const float* __restrict__ qw,  const float* __restrict__ qb,
    const float* __restrict__ kw,  const float* __restrict__ kb,
    float* __restrict__ msg, float* __restrict__ qn,
    float* __restrict__ kn,  float* __restrict__ outb,
    int n_nodes)
{
    __shared__ float sh[16 * DIM];          // 8 KB tile of h
    const int tid   = threadIdx.x;
    const int node0 = blockIdx.x * 16;

    // coalesced stage: 512 float4 slots, 2 per thread
    const float4* hsrc = (const float4*)(h + (size_t)node0 * DIM);
    float4* shv = (float4*)sh;
    int rows_valid = n_nodes - node0; if (rows_valid > 16) rows_valid = 16;
    int valid4 = rows_valid * (DIM / 4);
    for (int i = tid; i < 512; i += 256)
        shv[i] = (i < valid4) ? hsrc[i] : make_float4(0.f, 0.f, 0.f, 0.f);
    __syncthreads();

    const int lane  = tid & 31;
    const int wave  = tid >> 5;
    const int n0    = wave * 16;            // column tile
    const int rowA  = lane & 15;            // A-fragment row (M)
    const int khalf = (lane >> 4) * 2;      // upper half-wave carries K+2,K+3
    const int ncol  = n0 + (lane & 15);     // B-fragment column (N)

    const float* Wmat[4] = {Ww, qw, kw, WSw};
    const float* Bias[4] = {Wb, qb, kb, WSb};
    float*       Outp[4] = {msg, qn, kn, outb};

    const float2* arow = (const float2*)(sh + rowA * DIM);

    #pragma unroll
    for (int mtx = 0; mtx < 4; ++mtx) {
        const float* wr = Wmat[mtx] + (size_t)ncol * DIM;   // row of W == col of W^T
        v8f acc = {0.f, 0.f, 0.f, 0.f, 0.f, 0.f, 0.f, 0.f};
        #pragma unroll
        for (int k0 = 0; k0 < DIM; k0 += 4) {
            float2 a2 = arow[(k0 + khalf) >> 1];            // h[m][k], h[m][k+1]
            float2 b2 = *(const float2*)(wr + k0 + khalf);  // W[n][k], W[n][k+1]
            v2f a = {a2.x, a2.y};
            v2f b = {b2.x, b2.y};
            acc = __builtin_amdgcn_wmma_f32_16x16x4_f32(
                false, a, false, b, (short)0, acc, false, false);
        }
        const float bias = Bias[mtx][ncol];
        float* op = Outp[mtx];
        const int mbase = (lane >> 4) * 8;   // lanes 16-31 hold rows 8..15
        #pragma unroll
        for (int r = 0; r < 8; ++r) {
            int node = node0 + mbase + r;
            if (node < n_nodes)
                op[(size_t)node * DIM + ncol] = acc[r] + bias;
        }
    }
}

// ---------------------------------------------------------------------------
// 2. per-edge attention logits + segment max (one wave per edge)
// ---------------------------------------------------------------------------
__device__ __forceinline__ void atomicMaxFloat(float* addr, float val) {
    if (val >= 0.f) atomicMax((int*)addr, __float_as_int(val));
    else            atomicMin((unsigned int*)addr, __float_as_uint(val));
}

__global__ __launch_bounds__(256) void edge_scores_kernel(
    const float* __restrict__ qn, const float* __restrict__ kn,
    const int* __restrict__ src,  const int* __restrict__ dst,
    float* __restrict__ scores,   float* __restrict__ smax, int n_edges)
{
    int e    = blockIdx.x * 8 + (threadIdx.x >> 5);
    int lane = threadIdx.x & 31;
    if (e >= n_edges) return;
    int s = src[e], d = dst[e];
    float4 a = ((const float4*)(qn + (size_t)s * DIM))[lane];
    float4 b = ((const float4*)(kn + (size_t)d * DIM))[lane];
    float acc = a.x * b.x + a.y * b.y + a.z * b.z + a.w * b.w;
    #pragma unroll
    for (int off = 16; off > 0; off >>= 1)
        acc += __shfl_xor(acc, off, 32);
    if (lane == 0) {
        float sc = acc * 0.08838834764831845f;   // 1/sqrt(128)
        scores[e] = sc;
        atomicMaxFloat(smax + d, sc);
    }
}

// ---------------------------------------------------------------------------
// 3. exp(score - max) + segment sum
// ---------------------------------------------------------------------------
__global__ __launch_bounds__(256) void edge_exp_kernel(
    const float* __restrict__ scores, const int* __restrict__ dst,
    const float* __restrict__ smax,   float* __restrict__ evals,
    float* __restrict__ denom, int n_edges)
{
    int e = blockIdx.x * blockDim.x + threadIdx.x;
    if (e >= n_edges) return;
    int d = dst[e];
    float ev = expf(scores[e] - smax[d]);
    evals[e] = ev;
    atomicAdd(denom + d, ev);
}

// ---------------------------------------------------------------------------
// 4. alpha-weighted scatter of messages into out (one wave per edge)
// ---------------------------------------------------------------------------
__global__ __launch_bounds__(256) void edge_scatter_kernel(
    const float* __restrict__ evals, const float* __restrict__ denom,
    const float* __restrict__ msg,   const int* __restrict__ src,
    const int* __restrict__ dst,     float* __restrict__ out, int n_edges)
{
    int e    = blockIdx.x * 8 + (threadIdx.x >> 5);
    int lane = threadIdx.x & 31;
    if (e >= n_edges) return;
    int s = src[e], d = dst[e];
    float alpha = evals[e] / denom[d];
    const float* m = msg + (size_t)s * DIM;
    float*       o = out + (size_t)d * DIM;
    #pragma unroll
    for (int j = 0; j < 4; ++j) {
        int idx = lane + j * 32;
        atomicAdd(o + idx, alpha * m[idx]);
    }
}

// ---------------------------------------------------------------------------
// launch
// ---------------------------------------------------------------------------
extern "C" void kernel_launch(void* const* d_in, const int* in_sizes, int n_in,
                              void* d_out, int out_size, void* d_ws, size_t ws_size,
                              hipStream_t stream) {
    const float* h   = (const float*)d_in[0];
    const int*   src = (const int*)  d_in[1];
    const int*   dst = (const int*)  d_in[2];
    const float* Ww  = (const float*)d_in[3];
    const float* Wb  = (const float*)d_in[4];
    const float* WSw = (const float*)d_in[5];
    const float* WSb = (const float*)d_in[6];
    const float* qw  = (const float*)d_in[7];
    const float* qb  = (const float*)d_in[8];
    const float* kw  = (const float*)d_in[9];
    const float* kb  = (const float*)d_in[10];
    float* out = (float*)d_out;

    const int N = in_sizes[0] / DIM;
    const int E = in_sizes[1];

    // workspace carve (floats): 3 node matrices + 2 edge arrays + 2 node vecs
    float* ws     = (float*)d_ws;
    float* msg    = ws;
    float* qn     = msg    + (size_t)N * DIM;
    float* kn     = qn     + (size_t)N * DIM;
    float* scores = kn     + (size_t)N * DIM;
    float* evals  = scores + E;
    float* smax   = evals  + E;
    float* denom  = smax   + N;

    init_nodes<<<(N + 255) / 256, 256, 0, stream>>>(smax, denom, N);
    node_gemm4<<<(N + 15) / 16, 256, 0, stream>>>(h, Ww, Wb, WSw, WSb,
                                                  qw, qb, kw, kb,
                                                  msg, qn, kn, out, N);
    edge_scores_kernel<<<(E + 7) / 8, 256, 0, stream>>>(qn, kn, src, dst,
                                                        scores, smax, E);
    edge_exp_kernel<<<(E + 255) / 256, 256, 0, stream>>>(scores, dst, smax,
                                                         evals, denom, E);
    edge_scatter_kernel<<<(E + 7) / 8, 256, 0, stream>>>(evals, denom, msg,
                                                         src, dst, out, E);
}